// ConflictAwareResidualRouter_71949292142926
// MI455X (gfx1250) — compile-verified
//
#include <hip/hip_runtime.h>
#include <math.h>

// ---------------- problem constants (from reference) ----------------
#define NADP 4
#define BB   4
#define SS   2048
#define TT   (BB*SS)     // 8192 tokens
#define DD   4096
#define RH   64
#define HH   128
#define NC   6           // 2 + NADP logits
#define MB   64          // tokens per workgroup (phase 1)
#define KC   32          // K chunk staged in LDS
#define BLK  256         // 8 waves

typedef __attribute__((ext_vector_type(2))) float v2f;
typedef __attribute__((ext_vector_type(4))) float v4f;
typedef __attribute__((ext_vector_type(8))) float v8f;

// ---------------- LDS layout (floats), padded strides ----------------
#define HS_ST 36    // h tile row stride (144B, 16B aligned, conflict-free b64)
#define WST   136   // unified weight row stride (544B; halves hit disjoint banks)
#define FT_ST 65    // feat finish buffer row stride
#define HD_ST 129   // hid finish buffer row stride

#define SM_HS 0
#define SM_WR (MB*HS_ST)                 // 2304
#define SM_WG (SM_WR + KC*WST)           // 6656  (stage end = 11008)
#define SM_FT 0
#define SM_HD (MB*FT_ST)                 // 4160
#define SM_TOTAL (SM_HD + MB*HD_ST)      // 12416 floats = 49664 B

// ---------------- CDNA5 async global->LDS via inline asm ----------------
// ISA 15.18.3: GLOBAL_LOAD_ASYNC_TO_LDS_B128 (opcode 98), tracked by ASYNCcnt.
// vdst = VGPR holding LDS byte offset, vaddr = 64-bit global address, no SADDR.
#if __has_builtin(__builtin_amdgcn_s_wait_asynccnt)
#define USE_ASYNC_LDS 1
#endif

typedef __attribute__((address_space(3))) void as3void;

__device__ __forceinline__ void cp16(const float* g, float* l) {
#if USE_ASYNC_LDS
  unsigned lds_addr = (unsigned)(unsigned long long)(as3void*)l;
  asm volatile("global_load_async_to_lds_b128 %0, %1, off"
               :: "v"(lds_addr), "v"(g) : "memory");
#else
  *(v4f*)l = *(const v4f*)g;
#endif
}
__device__ __forceinline__ void cp_wait() {
#if USE_ASYNC_LDS
  __builtin_amdgcn_s_wait_asynccnt(0);
#endif
}

// =====================================================================
// Phase 1: per-token router gates via fused f32 WMMA GEMMs.
// Each WG: 64 tokens. 8 waves; wave = {tile 0..3} x {grp 0..1}.
//   grp0: rel N-tiles 0..3 + gate N-tiles 0..1   (6 WMMA tiles)
//   grp1: gate N-tiles 2..7                      (6 WMMA tiles)
// Inner loop is branch-free: per-wave job table of 6 LDS base offsets.
// =====================================================================
__global__ __launch_bounds__(BLK)
void router_gates_kernel(const float* __restrict__ h,
                         const float* __restrict__ conflict,
                         const float* __restrict__ rpw,
                         const float* __restrict__ rpb,
                         const float* __restrict__ rhw,
                         const float* __restrict__ rhb,
                         const float* __restrict__ gw1,
                         const float* __restrict__ gb1,
                         const float* __restrict__ gw2,
                         const float* __restrict__ gb2,
                         float* __restrict__ gates)
{
  __shared__ float sm[SM_TOTAL];

  const int tid  = threadIdx.x;
  const int wave = tid >> 5;
  const int lane = tid & 31;
  const int half = lane >> 4;    // WMMA lane half
  const int l16  = lane & 15;
  const int tile = wave & 3;     // token tile (16 rows)
  const int grp  = wave >> 2;    // N-split group
  const int tok0 = blockIdx.x * MB;

  // per-wave job table: 6 B-fragment base offsets (uniform per wave)
  int base[6];
  if (grp == 0) {
#pragma unroll
    for (int j = 0; j < 4; ++j) base[j] = SM_WR + j*16;   // rel N-tiles 0..3
    base[4] = SM_WG + 0;                                  // gate N-tiles 0..1
    base[5] = SM_WG + 16;
  } else {
#pragma unroll
    for (int j = 0; j < 6; ++j) base[j] = SM_WG + (j+2)*16; // gate N-tiles 2..7
  }
#pragma unroll
  for (int j = 0; j < 6; ++j) base[j] += half*2*WST + l16;  // fold lane terms
  const int aoff = SM_HS + (tile*16 + l16)*HS_ST + half*2;  // A-frag base

  v8f acc[6] = {};

  for (int k0 = 0; k0 < DD; k0 += KC) {
    __syncthreads();   // previous chunk fully consumed
    // ---- stage h[64 x 32], rel_proj_w[32 x 64], gate_w1[32 x 128] ----
    for (int c = tid; c < MB*KC/4; c += BLK) {          // 512 x 16B
      int r = c >> 3, col = (c & 7) * 4;
      cp16(h + (size_t)(tok0 + r)*DD + k0 + col, &sm[SM_HS + r*HS_ST + col]);
    }
    for (int c = tid; c < KC*RH/4; c += BLK) {          // 512 x 16B
      int r = c >> 4, col = (c & 15) * 4;
      cp16(rpw + (size_t)(k0 + r)*RH + col, &sm[SM_WR + r*WST + col]);
    }
    for (int c = tid; c < KC*HH/4; c += BLK) {          // 1024 x 16B
      int r = c >> 5, col = (c & 31) * 4;
      cp16(gw1 + (size_t)(k0 + r)*HH + col, &sm[SM_WG + r*WST + col]);
    }
    cp_wait();
    __syncthreads();

    // ---- 8 x v_wmma_f32_16x16x4_f32 steps per chunk, branch-free ----
#pragma unroll
    for (int kk = 0; kk < KC; kk += 4) {
      // A frag 16x4: lanes<16 hold K=0,1 ; lanes>=16 hold K=2,3
      v2f a = *(const v2f*)&sm[aoff + kk];
#pragma unroll
      for (int j = 0; j < 6; ++j) {
        const int o = base[j] + kk*WST;     // kk*WST is a fold-time constant
        v2f b; b.x = sm[o]; b.y = sm[o + WST];
        acc[j] = __builtin_amdgcn_wmma_f32_16x16x4_f32(
            false, a, false, b, (short)0, acc[j], false, false);
      }
    }
  }

  // ---- spill accumulators to finish buffers (C layout: row=i+8*half) ----
  __syncthreads();
  if (grp == 0) {
#pragma unroll
    for (int j = 0; j < 4; ++j)
#pragma unroll
      for (int i = 0; i < 8; ++i) {
        int row = tile*16 + half*8 + i, col = j*16 + l16;
        sm[SM_FT + row*FT_ST + col] = fmaxf(acc[j][i] + rpb[col], 0.f);
      }
#pragma unroll
    for (int j = 4; j < 6; ++j)
#pragma unroll
      for (int i = 0; i < 8; ++i) {
        int row = tile*16 + half*8 + i, col = (j-4)*16 + l16;
        sm[SM_HD + row*HD_ST + col] = acc[j][i];
      }
  } else {
#pragma unroll
    for (int j = 0; j < 6; ++j)
#pragma unroll
      for (int i = 0; i < 8; ++i) {
        int row = tile*16 + half*8 + i, col = (j+2)*16 + l16;
        sm[SM_HD + row*HD_ST + col] = acc[j][i];
      }
  }
  __syncthreads();

  // ---- per-token finish: rel, rank-8 correction, logits, top-2, softmax ----
  if (tid < MB) {
    const int token = tok0 + tid;
    float racc[NADP];
#pragma unroll
    for (int a = 0; a < NADP; ++a) racc[a] = rhb[a];
    const float* fr = &sm[SM_FT + tid*FT_ST];
    for (int c = 0; c < RH; ++c) {
      float f = fr[c];
#pragma unroll
      for (int a = 0; a < NADP; ++a) racc[a] += f * rhw[c*NADP + a];
    }
    float relv[NADP], conf[NADP];
#pragma unroll
    for (int a = 0; a < NADP; ++a) {
      relv[a] = 1.f / (1.f + expf(-racc[a]));
      conf[a] = conflict[(size_t)token*NADP + a];
    }
    float lg[NC];
#pragma unroll
    for (int c = 0; c < NC; ++c) lg[c] = 0.f;
    const float* hr = &sm[SM_HD + tid*HD_ST];
    for (int x = 0; x < HH; ++x) {
      float v = hr[x] + gb1[x];
#pragma unroll
      for (int j = 0; j < NADP; ++j) v += relv[j] * gw1[(size_t)(DD + j)*HH + x];
#pragma unroll
      for (int j = 0; j < NADP; ++j) v += conf[j] * gw1[(size_t)(DD + NADP + j)*HH + x];
      v = fmaxf(v, 0.f);
#pragma unroll
      for (int c = 0; c < NC; ++c) lg[c] += v * gw2[x*NC + c];
    }
#pragma unroll
    for (int c = 0; c < NC; ++c) lg[c] += gb2[c];   // TEMPERATURE == 1

    // top-2 of adapter logits lg[2..5] (ties -> lower index, like lax.top_k)
    int i1 = 2;
#pragma unroll
    for (int a = 3; a < NC; ++a) if (lg[a] > lg[i1]) i1 = a;
    int i2 = -1;
#pragma unroll
    for (int a = 2; a < NC; ++a)
      if (a != i1 && (i2 < 0 || lg[a] > lg[i2])) i2 = a;
#pragma unroll
    for (int a = 2; a < NC; ++a)
      if (a != i1 && a != i2) lg[a] = -INFINITY;

    float m = lg[0];
#pragma unroll
    for (int c = 1; c < NC; ++c) m = fmaxf(m, lg[c]);
    float e[NC], s = 0.f;
#pragma unroll
    for (int c = 0; c < NC; ++c) { e[c] = expf(lg[c] - m); s += e[c]; }
    float inv = 1.f / s;
    float* gp = gates + (size_t)token * 5;
    gp[0] = e[1] * inv;                       // static_delta gate
#pragma unroll
    for (int a = 0; a < NADP; ++a) gp[1 + a] = e[2 + a] * inv;
  }
}

// =====================================================================
// Phase 2: streaming combine. 768 MB of traffic -> nontemporal b128.
// =====================================================================
__global__ __launch_bounds__(256)
void router_combine_kernel(const float* __restrict__ sd,
                           const float* __restrict__ res,
                           const float* __restrict__ gates,
                           float* __restrict__ out)
{
  const size_t TD4 = (size_t)TT * (DD/4);   // 8388608 float4s
  const v4f* sd4  = (const v4f*)sd;
  const v4f* rs4  = (const v4f*)res;
  v4f*       out4 = (v4f*)out;
  size_t stride = (size_t)gridDim.x * blockDim.x;
  for (size_t v = (size_t)blockIdx.x * blockDim.x + threadIdx.x; v < TD4; v += stride) {
    size_t tok = v >> 10;                    // DD/4 == 1024
    const float* g = gates + tok * 5;
    float g1 = g[0], a0 = g[1], a1 = g[2], a2 = g[3], a3 = g[4];
    v4f s  = __builtin_nontemporal_load(sd4 + v);
    v4f r0 = __builtin_nontemporal_load(rs4 + v);
    v4f r1 = __builtin_nontemporal_load(rs4 + TD4 + v);
    v4f r2 = __builtin_nontemporal_load(rs4 + 2*TD4 + v);
    v4f r3 = __builtin_nontemporal_load(rs4 + 3*TD4 + v);
    v4f o = g1*s + a0*r0 + a1*r1 + a2*r2 + a3*r3;
    __builtin_nontemporal_store(o, out4 + v);
  }
}

extern "C" void kernel_launch(void* const* d_in, const int* in_sizes, int n_in,
                              void* d_out, int out_size, void* d_ws, size_t ws_size,
                              hipStream_t stream) {
  (void)in_sizes; (void)n_in; (void)out_size; (void)ws_size;
  const float* h    = (const float*)d_in[0];
  const float* sd   = (const float*)d_in[1];
  const float* res  = (const float*)d_in[2];
  const float* conf = (const float*)d_in[3];
  const float* rpw  = (const float*)d_in[4];
  const float* rpb  = (const float*)d_in[5];
  const float* rhw  = (const float*)d_in[6];
  const float* rhb  = (const float*)d_in[7];
  const float* gw1  = (const float*)d_in[8];
  const float* gb1  = (const float*)d_in[9];
  const float* gw2  = (const float*)d_in[10];
  const float* gb2  = (const float*)d_in[11];
  float* gates = (float*)d_ws;              // 8192*5 floats = 160 KB
  float* out   = (float*)d_out;

  router_gates_kernel<<<TT/MB, BLK, 0, stream>>>(
      h, conf, rpw, rpb, rhw, rhb, gw1, gb1, gw2, gb2, gates);
  router_combine_kernel<<<8192, 256, 0, stream>>>(sd, res, gates, out);
}